// Attention2_55370718380196
// MI455X (gfx1250) — compile-verified
//
#include <hip/hip_runtime.h>
#include <hip/hip_bf16.h>

// ---------------------------------------------------------------------------
// Problem (after algebraic reduction of the reference):
//   per pair p in [0,256):   x_p = x[p] : [512,512] f32 (row-major)
//     q0  = x_p[0,:] @ Wq^T                     (one row of Q)
//     kq  = q0 @ Wk                             (fold K projection into q)
//     s_m = (x_p[m,:] . kq) / sqrt(512)
//     prob = softmax(s)
//     xv  = sum_m prob[m] * x_p[m,:]            (fold V projection out)
//     out_p = xv @ Wv^T
// Batched across pairs, steps 1,2,5 are [256x512]x[512x512] GEMMs -> WMMA f32.
// Step 3/4 is a bandwidth-bound pass over x (268 MB ~= HBM roofline).
// ---------------------------------------------------------------------------

typedef float v2f __attribute__((ext_vector_type(2)));
typedef float v8f __attribute__((ext_vector_type(8)));

#define DIM   512
#define PAIRS 256
// 1 / sqrt(512)
#define INV_TEMP 0.04419417382415922f

// ----------------------------------------------------------------------------
// C[256,512] = A[256,512] x op(W), using V_WMMA_F32_16X16X4_F32.
//   TRANSB=true :  C[n,o] = sum_d A[n,d] * W[o,d]   (W row-major [512,512])
//   TRANSB=false:  C[n,o] = sum_d A[n,d] * W[d,o]
// A row stride = ldA (x0 rows are 512*512 apart; ws matrices are dense).
// One 16x16 output tile per wave, K stepped by 4; 8 waves / block, 64 blocks.
// Fragment layouts per CDNA5 ISA 7.12.2:
//   A 16x4 : lanes 0-15 M=lane {K0,K1}; lanes 16-31 M=lane-16 {K2,K3}
//   B 4x16 : lanes 0-15 N=lane rows {K0,K1}; lanes 16-31 rows {K2,K3}
//   C/D    : VGPR r -> M = r (lanes 0-15) / r+8 (lanes 16-31), N = lane&15
// ----------------------------------------------------------------------------
template <bool TRANSB>
__global__ __launch_bounds__(256) void gemm256x512_wmma(
    const float* __restrict__ A, long ldA,
    const float* __restrict__ W,
    float* __restrict__ C)
{
    const int wave = threadIdx.x >> 5;
    const int lane = threadIdx.x & 31;
    const int hf   = lane >> 4;      // which half of the wave
    const int lrow = lane & 15;

    const int tile = blockIdx.x * 8 + wave;   // 512 tiles total
    const int tM   = tile & 15;               // 16 row tiles (M = 256)
    const int tN   = tile >> 4;               // 32 col tiles (N = 512)
    const int rowBase = tM << 4;
    const int colBase = tN << 4;

    const float* arow = A + (size_t)(rowBase + lrow) * ldA;

    v8f acc = {};
    for (int k0 = 0; k0 < DIM; k0 += 4) {
        const int ka = k0 + 2 * hf;

        v2f a;
        a.x = arow[ka];
        a.y = arow[ka + 1];

        v2f b;
        if (TRANSB) {
            const float* wr = W + (size_t)(colBase + lrow) * DIM + ka;
            b.x = wr[0];
            b.y = wr[1];
        } else {
            b.x = W[(size_t)ka * DIM + colBase + lrow];
            b.y = W[(size_t)(ka + 1) * DIM + colBase + lrow];
        }

        // (neg_a, A, neg_b, B, c_mod, C, reuse_a, reuse_b)
        acc = __builtin_amdgcn_wmma_f32_16x16x4_f32(
            false, a, false, b, (short)0, acc, false, false);
    }

#pragma unroll
    for (int r = 0; r < 8; ++r) {
        C[(size_t)(rowBase + r + 8 * hf) * DIM + colBase + lrow] = acc[r];
    }
}

// ----------------------------------------------------------------------------
// Per-pair attention core (bandwidth phase). One block (8 wave32) per pair.
//   phase 1: scores[m] = (x_p[m,:] . kq) / temp   (float4 coalesced, wave
//            shuffle reduction; 64 rows per wave)
//   phase 2: block softmax over 512 scores in LDS
//   phase 3: xv[d] = sum_m prob[m] * x_p[m,d]     (x re-read hits L2)
// ----------------------------------------------------------------------------
__global__ __launch_bounds__(256) void attn_core(
    const float* __restrict__ x,
    const float* __restrict__ kq,
    float* __restrict__ xv)
{
    __shared__ __align__(16) float kq_s[DIM];
    __shared__ float sc[DIM];
    __shared__ float red[256];

    const int tid  = threadIdx.x;
    const int pair = blockIdx.x;
    const float* xp = x + (size_t)pair * DIM * DIM;

    // stage kq into LDS
    kq_s[tid]       = kq[(size_t)pair * DIM + tid];
    kq_s[tid + 256] = kq[(size_t)pair * DIM + tid + 256];
    __syncthreads();

    // ---- scores ----
    const int wave = tid >> 5;
    const int lane = tid & 31;
    const float4* kq4 = (const float4*)kq_s;
    for (int m = wave * 64; m < (wave + 1) * 64; ++m) {
        const float4* xr = (const float4*)(xp + (size_t)m * DIM);
        float acc = 0.0f;
#pragma unroll
        for (int c = 0; c < 4; ++c) {
            float4 xe = xr[lane + 32 * c];
            float4 ke = kq4[lane + 32 * c];
            acc = fmaf(xe.x, ke.x, acc);
            acc = fmaf(xe.y, ke.y, acc);
            acc = fmaf(xe.z, ke.z, acc);
            acc = fmaf(xe.w, ke.w, acc);
        }
#pragma unroll
        for (int off = 16; off > 0; off >>= 1)
            acc += __shfl_down(acc, off, 32);
        if (lane == 0) sc[m] = acc * INV_TEMP;
    }
    __syncthreads();

    // ---- softmax over 512 entries ----
    red[tid] = fmaxf(sc[tid], sc[tid + 256]);
    __syncthreads();
    for (int s = 128; s > 0; s >>= 1) {
        if (tid < s) red[tid] = fmaxf(red[tid], red[tid + s]);
        __syncthreads();
    }
    const float mx = red[0];
    __syncthreads();

    const float e0 = __expf(sc[tid] - mx);
    const float e1 = __expf(sc[tid + 256] - mx);
    red[tid] = e0 + e1;
    __syncthreads();
    for (int s = 128; s > 0; s >>= 1) {
        if (tid < s) red[tid] += red[tid + s];
        __syncthreads();
    }
    const float inv = 1.0f / red[0];
    __syncthreads();
    sc[tid]       = e0 * inv;
    sc[tid + 256] = e1 * inv;
    __syncthreads();

    // ---- xv = x^T . prob  (columns tid, tid+256) ----
    float a0 = 0.0f, a1 = 0.0f;
    for (int m = 0; m < DIM; ++m) {
        const float pm = sc[m];
        a0 = fmaf(pm, xp[(size_t)m * DIM + tid], a0);
        a1 = fmaf(pm, xp[(size_t)m * DIM + tid + 256], a1);
    }
    xv[(size_t)pair * DIM + tid]       = a0;
    xv[(size_t)pair * DIM + tid + 256] = a1;
}

// ----------------------------------------------------------------------------
extern "C" void kernel_launch(void* const* d_in, const int* in_sizes, int n_in,
                              void* d_out, int out_size, void* d_ws, size_t ws_size,
                              hipStream_t stream) {
    const float* x  = (const float*)d_in[0];   // [8,32,512,512]
    const float* Wq = (const float*)d_in[1];   // [512,512]
    const float* Wk = (const float*)d_in[2];
    const float* Wv = (const float*)d_in[3];
    float* out = (float*)d_out;                // [256,512]

    float* q0 = (float*)d_ws;                  // [256,512]
    float* kq = q0 + (size_t)PAIRS * DIM;      // [256,512]
    float* xv = q0;                            // reuse q0's buffer (dead by then)

    // q0 = X0 @ Wq^T   (X0 rows are x[pair][0][:], stride 512*512)
    gemm256x512_wmma<true><<<64, 256, 0, stream>>>(x, (long)DIM * DIM, Wq, q0);
    // kq = q0 @ Wk
    gemm256x512_wmma<false><<<64, 256, 0, stream>>>(q0, DIM, Wk, kq);
    // scores + softmax + xv (bandwidth phase: one HBM read of x)
    attn_core<<<PAIRS, 256, 0, stream>>>(x, kq, xv);
    // out = xv @ Wv^T
    gemm256x512_wmma<true><<<64, 256, 0, stream>>>(xv, DIM, Wv, out);
}